// SurpriseGate_11433202942763
// MI455X (gfx1250) — compile-verified
//
#include <hip/hip_runtime.h>

// Problem constants (match reference)
#define B_    8
#define S_    4096
#define D_    1024
#define G_    64
#define HSEQ_ 2048

typedef __attribute__((ext_vector_type(2))) float v2f;
typedef __attribute__((ext_vector_type(4))) float v4f;
typedef __attribute__((ext_vector_type(8))) float v8f;

// ---------------- workspace layout (in floats) ----------------
// [0,           B*D)   q_probe accumulators (sum over H_SEQ)
// [QEND,        +B)    k_surprise[b]
// [..,          +B)    v_surprise[b]
// [..,          +B*G)  k_gate[b][g]
// [..,          +B*G)  v_gate[b][g]
#define QOFF   0
#define KSOFF  (B_ * D_)
#define VSOFF  (KSOFF + B_)
#define KGOFF  (VSOFF + B_)
#define VGOFF  (KGOFF + B_ * G_)

__device__ __forceinline__ float sigmoidf_(float x) {
    return 1.0f / (1.0f + __expf(-x));
}

// ---------------------------------------------------------------------------
// Kernel 0: zero the q_probe accumulators (ws is poisoned, not re-zeroed)
// ---------------------------------------------------------------------------
__global__ void sg_zero_q(float* __restrict__ ws) {
    ws[QOFF + blockIdx.x * 256 + threadIdx.x] = 0.0f;
}

// ---------------------------------------------------------------------------
// Kernel 1: q_probe sums = sum_s h[b,s,:]   (scaled by 1/H_SEQ at use site)
// grid (64, B), block 256; float4 coalesced rows, atomicAdd accumulation
// ---------------------------------------------------------------------------
__global__ void sg_reduce_h(const float* __restrict__ h, float* __restrict__ ws) {
    const int b     = blockIdx.y;
    const int chunk = blockIdx.x;               // 64 chunks of 32 rows
    const int d     = threadIdx.x * 4;          // 256 threads * 4 = 1024 = D
    const int rows  = HSEQ_ / 64;
    const float* hp = h + ((size_t)b * HSEQ_ + (size_t)chunk * rows) * D_ + d;
    float ax = 0.f, ay = 0.f, az = 0.f, aw = 0.f;
    for (int s = 0; s < rows; ++s) {
        v4f x = *(const v4f*)(hp + (size_t)s * D_);
        ax += x.x; ay += x.y; az += x.z; aw += x.w;
    }
    float* q = ws + QOFF + b * D_ + d;
    atomicAdd(q + 0, ax);
    atomicAdd(q + 1, ay);
    atomicAdd(q + 2, az);
    atomicAdd(q + 3, aw);
}

// ---------------------------------------------------------------------------
// Kernel 2: probe attention + surprise, one block per (batch, K-or-V).
// Scores via V_WMMA_F32_16X16X4_F32:
//   A (16x4 f32): lane<16 -> M=lane, K={4i,4i+1}; lane>=16 -> M=lane-16, K={4i+2,4i+3}
//   B (4x16)    : q broadcast across all 16 columns -> D columns all equal score
// 4 waves handle the 4 groups of 16 rows; C accumulates over D/4 = 256 steps.
// ---------------------------------------------------------------------------
__global__ void sg_probe(const float* __restrict__ K_curr,
                         const float* __restrict__ V_curr,
                         const int*   __restrict__ active_idx,
                         float*       __restrict__ ws) {
    __shared__ float qs[D_];
    __shared__ float sco[G_];
    __shared__ float attn[G_];
    __shared__ int   rows[G_];
    __shared__ float red[128];

    const int b   = blockIdx.x;
    const int mat = blockIdx.y;                 // 0 = K, 1 = V
    const int tid = threadIdx.x;                // 128 threads = 4 waves
    const float* Xb = (mat ? V_curr : K_curr) + (size_t)b * S_ * D_;

    const float invH = 1.0f / (float)HSEQ_;
    for (int d = tid; d < D_; d += 128) qs[d] = ws[QOFF + b * D_ + d] * invH;
    if (tid < G_) rows[tid] = active_idx[b * G_ + tid];
    __syncthreads();

    // ---- WMMA score GEMV: scores[g] = q . X[g,:] ----
    const int wave = tid >> 5;
    const int lane = tid & 31;
    const int m    = lane & 15;
    const int hi   = lane >> 4;                 // 0: K pair {0,1}, 1: K pair {2,3}
    const float* rp = Xb + (size_t)rows[16 * wave + m] * D_ + 2 * hi;
    const float* qp = qs + 2 * hi;
    v8f c = {};
    for (int i = 0; i < D_ / 4; ++i) {
        v2f a  = *(const v2f*)(rp);
        v2f qv = *(const v2f*)(qp);
        rp += 4; qp += 4;
        c = __builtin_amdgcn_wmma_f32_16x16x4_f32(
                /*neg_a=*/false, a, /*neg_b=*/false, qv,
                /*c_mod=*/(short)0, c, /*reuse_a=*/false, /*reuse_b=*/false);
    }
    // C/D layout: lane L<16 holds D[M=v][N=L] in c[v]; lane L>=16 holds M=v+8.
    // All N columns are identical (B was column-broadcast), so use lanes 0/16.
    if (lane == 0)
        for (int v = 0; v < 8; ++v) sco[16 * wave + v] = c[v];
    if (lane == 16)
        for (int v = 0; v < 8; ++v) sco[16 * wave + 8 + v] = c[v];
    __syncthreads();

    // ---- softmax over G=64 (tiny, serial on one lane) ----
    if (tid == 0) {
        const float scale = 0.03125f;           // D^-0.5 = 1/32
        float mx = -3.0e38f;
        for (int g = 0; g < G_; ++g) {
            float s = sco[g] * scale;
            sco[g] = s;
            mx = s > mx ? s : mx;
        }
        float sum = 0.f;
        for (int g = 0; g < G_; ++g) { float e = __expf(sco[g] - mx); attn[g] = e; sum += e; }
        float inv = 1.0f / sum;
        for (int g = 0; g < G_; ++g) attn[g] *= inv;
    }
    __syncthreads();

    // ---- pred = attn @ X ; surprise = mean((pred - q)^2) ----
    float err = 0.f;
    for (int d = tid; d < D_; d += 128) {
        float p = 0.f;
        for (int g = 0; g < G_; ++g)
            p += attn[g] * Xb[(size_t)rows[g] * D_ + d];
        float diff = p - qs[d];
        err += diff * diff;
    }
    red[tid] = err;
    __syncthreads();
    for (int s = 64; s > 0; s >>= 1) {
        if (tid < s) red[tid] += red[tid + s];
        __syncthreads();
    }
    if (tid == 0)
        ws[(mat ? VSOFF : KSOFF) + b] = red[0] * (1.0f / (float)D_);
}

// ---------------------------------------------------------------------------
// Kernel 3: momentum update + gate projections (1 block, 512 threads)
// ---------------------------------------------------------------------------
__global__ void sg_gates(const float* __restrict__ momentum,
                         const float* __restrict__ Wk, const float* __restrict__ bk,
                         const float* __restrict__ Wv, const float* __restrict__ bv,
                         const float* __restrict__ logit_eta,
                         const float* __restrict__ alpha_logit,
                         float* __restrict__ ws,
                         float* __restrict__ out_momentum) {
    const int t = threadIdx.x;                  // 512 = B*G
    const int b = t >> 6;
    const int g = t & (G_ - 1);
    const float ks = ws[KSOFF + b];
    const float vs = ws[VSOFF + b];
    const float alpha = sigmoidf_(alpha_logit[0]);
    const float eta   = sigmoidf_(logit_eta[0]);
    const float combined = alpha * ks + (1.0f - alpha) * vs;
    const float nm = eta * momentum[b] + (1.0f - eta) * combined;
    // k_in = [k_surprise, new_momentum] @ Wk[g,:] + bk[g]
    const float kg = sigmoidf_(ks * Wk[2 * g] + nm * Wk[2 * g + 1] + bk[g]);
    const float vg = sigmoidf_(vs * Wv[2 * g] + nm * Wv[2 * g + 1] + bv[g]);
    ws[KGOFF + t] = kg;
    ws[VGOFF + t] = vg;
    if (g == 0) out_momentum[b] = nm;
}

// ---------------------------------------------------------------------------
// Kernel 4: bulk streaming copy K_curr->out0, V_curr->out1 (non-temporal b128)
// ---------------------------------------------------------------------------
__global__ void sg_copy(const v4f* __restrict__ Kc, const v4f* __restrict__ Vc,
                        v4f* __restrict__ oK, v4f* __restrict__ oV) {
    const size_t i = (size_t)blockIdx.x * blockDim.x + threadIdx.x;
    v4f k = __builtin_nontemporal_load(Kc + i);
    v4f v = __builtin_nontemporal_load(Vc + i);
    __builtin_nontemporal_store(k, oK + i);
    __builtin_nontemporal_store(v, oV + i);
}

// ---------------------------------------------------------------------------
// Kernel 5: overwrite active rows with gated blend.
// grid (B*G, 2), block 256 (one float4 per thread covers D=1024)
// ---------------------------------------------------------------------------
__global__ void sg_fixup(const float* __restrict__ Kc, const float* __restrict__ Vc,
                         const float* __restrict__ Kp, const float* __restrict__ Vp,
                         const int*   __restrict__ active_idx,
                         const float* __restrict__ ws,
                         float*       __restrict__ out) {
    const int bg  = blockIdx.x;                 // b*G + g
    const int mat = blockIdx.y;
    const int b   = bg >> 6;
    const int row = active_idx[bg];
    const float gate = ws[(mat ? VGOFF : KGOFF) + bg];
    const size_t roff = ((size_t)b * S_ + (size_t)row) * D_;
    const float* curr = (mat ? Vc : Kc) + roff;
    const float* prev = (mat ? Vp : Kp) + roff;
    float* o = out + (size_t)mat * B_ * S_ * D_ + roff;
    const int d = threadIdx.x * 4;
    v4f c = *(const v4f*)(curr + d);
    v4f p = *(const v4f*)(prev + d);
    v4f r = gate * c + (1.0f - gate) * p;
    *(v4f*)(o + d) = r;
}

// ---------------------------------------------------------------------------
extern "C" void kernel_launch(void* const* d_in, const int* in_sizes, int n_in,
                              void* d_out, int out_size, void* d_ws, size_t ws_size,
                              hipStream_t stream) {
    const float* K_curr      = (const float*)d_in[0];
    const float* V_curr      = (const float*)d_in[1];
    const float* K_prev      = (const float*)d_in[2];
    const float* V_prev      = (const float*)d_in[3];
    const float* h           = (const float*)d_in[4];
    const float* momentum    = (const float*)d_in[5];
    const int*   active_idx  = (const int*)d_in[6];
    const float* Wk          = (const float*)d_in[7];
    const float* bk          = (const float*)d_in[8];
    const float* Wv          = (const float*)d_in[9];
    const float* bv          = (const float*)d_in[10];
    const float* logit_eta   = (const float*)d_in[11];
    const float* alpha_logit = (const float*)d_in[12];

    float* out = (float*)d_out;
    float* ws  = (float*)d_ws;
    float* out_momentum = out + 2 * (size_t)B_ * S_ * D_;

    // 0) zero q_probe accumulators (B*D = 8192 floats)
    sg_zero_q<<<(B_ * D_) / 256, 256, 0, stream>>>(ws);

    // 1) q_probe sums over h
    sg_reduce_h<<<dim3(64, B_), 256, 0, stream>>>(h, ws);

    // 2) probe attention + surprise (WMMA scores), per (batch, K/V)
    sg_probe<<<dim3(B_, 2), 128, 0, stream>>>(K_curr, V_curr, active_idx, ws);

    // 3) momentum + gates
    sg_gates<<<1, B_ * G_, 0, stream>>>(momentum, Wk, bk, Wv, bv,
                                        logit_eta, alpha_logit, ws, out_momentum);

    // 4) bulk copy (bandwidth-bound: 512 MB stream)
    {
        const size_t n4 = (size_t)B_ * S_ * D_ / 4;   // 8,388,608 float4
        sg_copy<<<(unsigned)(n4 / 256), 256, 0, stream>>>(
            (const v4f*)K_curr, (const v4f*)V_curr,
            (v4f*)out, (v4f*)(out + (size_t)B_ * S_ * D_));
    }

    // 5) gated blend on active rows (overwrites copied rows)
    sg_fixup<<<dim3(B_ * G_, 2), 256, 0, stream>>>(
        K_curr, V_curr, K_prev, V_prev, active_idx, ws, out);
}